// SimMaxLoss_18528488915443
// MI455X (gfx1250) — compile-verified
//
#include <hip/hip_runtime.h>
#include <math.h>
#include <stdint.h>

typedef __attribute__((ext_vector_type(2))) float v2f;
typedef __attribute__((ext_vector_type(4))) float v4f;
typedef __attribute__((ext_vector_type(8))) float v8f;

#define C_DIM   512
#define NN      8192
#define BT      64      // block tile (M and N)
#define KT      64      // K tile staged in LDS
#define LDSD    68      // padded LDS row stride (floats): conflict-free frag reads
#define ALPHA   0.25f

// ---------------- Kernel 1: row L2 norms + write normalized x ----------------
__global__ __launch_bounds__(256) void rownorm_kernel(const float* __restrict__ e,
                                                      float* __restrict__ xn, int C) {
    __shared__ float red[256];
    const int row = blockIdx.x;
    const float* rp = e + (size_t)row * C;
    float s = 0.f;
    for (int c = threadIdx.x; c < C; c += 256) { float v = rp[c]; s += v * v; }
    red[threadIdx.x] = s;
    __syncthreads();
    for (int off = 128; off > 0; off >>= 1) {
        if (threadIdx.x < off) red[threadIdx.x] += red[threadIdx.x + off];
        __syncthreads();
    }
    const float rinv = 1.0f / fmaxf(sqrtf(red[0]), 1e-12f);
    float* op = xn + (size_t)row * C;
    for (int c = threadIdx.x; c < C; c += 256) op[c] = rp[c] * rinv;
}

// Stage one 64x64 fp32 tile (rows rowBase.., K-slice kb..) into LDS using the
// CDNA5 async global->LDS path (no VGPR round trip, tracked by ASYNCcnt).
// Each thread issues 4 x b128 per tile => 8 async ops per thread per K-chunk (A+B).
__device__ __forceinline__ void async_stage_tile(const float* __restrict__ gbase,
                                                 int rowBase, int C, int kb,
                                                 float* ldsTile, int tid) {
    #pragma unroll
    for (int it = 0; it < 4; ++it) {
        const int i  = it * 256 + tid;
        const int r  = i >> 4;            // row within tile
        const int c4 = (i & 15) << 2;     // float4 column
        const unsigned voff = (unsigned)((((size_t)(rowBase + r)) * C + kb + c4) * 4);
        const unsigned ldsa = (unsigned)(uintptr_t)(ldsTile + r * LDSD + c4);
        asm volatile("global_load_async_to_lds_b128 %0, %1, %2"
                     :: "v"(ldsa), "v"(voff), "s"(gbase)
                     : "memory");
    }
}

// ---------------- Kernel 2: sim = clamp(x x^T) via V_WMMA_F32_16X16X4_F32 ----
// 64x64 block per workgroup, 8 wave32s; wave w owns a 16x32 sub-tile.
// Double-buffered LDS with async prefetch of chunk k+1 overlapping WMMAs on k.
__global__ __launch_bounds__(256) void simgemm_kernel(const float* __restrict__ xn,
                                                      float* __restrict__ sim,
                                                      int N, int C) {
    __shared__ __align__(16) float As[2][BT * LDSD];
    __shared__ __align__(16) float Bs[2][BT * LDSD];
    const int tid    = threadIdx.x;
    const int lane   = tid & 31;
    const int wave   = tid >> 5;
    const int mt     = wave >> 1;
    const int ntBase = (wave & 1) * 32;
    const int l16    = lane & 15;
    const int hi     = lane >> 4;
    const int bm = blockIdx.y, bn = blockIdx.x;

    v8f acc0 = {};
    v8f acc1 = {};

    const int NCHUNK = C / KT;   // 8

    // Prologue: stage chunk 0 into buffer 0 (8 async ops per thread in flight).
    async_stage_tile(xn, bm * BT, C, 0, &As[0][0], tid);
    async_stage_tile(xn, bn * BT, C, 0, &Bs[0][0], tid);

    int buf = 0;
    for (int ck = 0; ck < NCHUNK; ++ck) {
        const bool havePrefetch = (ck + 1 < NCHUNK);
        if (havePrefetch) {
            // Prefetch next chunk into the other buffer. Safe: the barrier at
            // the end of the previous iteration guarantees nobody still reads it.
            const int kb = (ck + 1) * KT;
            async_stage_tile(xn, bm * BT, C, kb, &As[buf ^ 1][0], tid);
            async_stage_tile(xn, bn * BT, C, kb, &Bs[buf ^ 1][0], tid);
            // 16 outstanding now; wait until only the 8 prefetch ops remain,
            // i.e. my 8 current-chunk ops have landed in LDS.
            asm volatile("s_wait_asynccnt 0x8" ::: "memory");
        } else {
            asm volatile("s_wait_asynccnt 0x0" ::: "memory");
        }
        __syncthreads();   // all waves' current-chunk LDS writes visible

        const float* Ab = &As[buf][0];
        const float* Bb = &Bs[buf][0];
        #pragma unroll
        for (int kk = 0; kk < KT; kk += 4) {
            const int c = kk + 2 * hi;   // lanes 0-15: K0/K1; lanes 16-31: K2/K3
            v2f a  = *(const v2f*)(&Ab[(mt * 16 + l16) * LDSD + c]);
            v2f b0 = *(const v2f*)(&Bb[(ntBase      + l16) * LDSD + c]);
            v2f b1 = *(const v2f*)(&Bb[(ntBase + 16 + l16) * LDSD + c]);
            acc0 = __builtin_amdgcn_wmma_f32_16x16x4_f32(false, a, false, b0,
                                                         (short)0, acc0, false, false);
            acc1 = __builtin_amdgcn_wmma_f32_16x16x4_f32(false, a, false, b1,
                                                         (short)0, acc1, false, false);
        }
        __syncthreads();   // everyone done reading buf before it is prefetched
        buf ^= 1;
    }

    // C/D layout: VGPR v -> M = v (lanes 0-15) or v+8 (lanes 16-31); N = lane%16
    const int rowBlock = bm * BT, colBlock = bn * BT;
    #pragma unroll
    for (int v = 0; v < 8; ++v) {
        const int r  = rowBlock + mt * 16 + v + 8 * hi;
        const int c0 = colBlock + ntBase + l16;
        float x0 = fminf(fmaxf(acc0[v], 0.0005f), 0.9995f);
        float x1 = fminf(fmaxf(acc1[v], 0.0005f), 0.9995f);
        sim[(size_t)r * N + c0]      = x0;
        sim[(size_t)r * N + c0 + 16] = x1;
    }
}

// ---------------- Kernel 3: per-row bitonic sort (desc) + weighted -log sum --
__global__ __launch_bounds__(256) void rowsort_kernel(const float* __restrict__ sim,
                                                      float* __restrict__ rowsum, int N) {
    __shared__ float s[NN];
    __shared__ float red[256];
    const int tid = threadIdx.x;
    const int row = blockIdx.x;
    const float* rp = sim + (size_t)row * N;

    for (int k = tid; k < NN / 4; k += 256) {
        v4f v = *(const v4f*)(rp + 4 * k);
        *(v4f*)(&s[4 * k]) = v;
    }
    __syncthreads();

    // Bitonic sort, descending overall.
    for (int size = 2; size <= NN; size <<= 1) {
        for (int stride = size >> 1; stride > 0; stride >>= 1) {
            for (int t = tid; t < NN / 2; t += 256) {
                const int low = t & (stride - 1);
                const int i = (t << 1) - low;
                const int j = i + stride;
                const float a = s[i], b = s[j];
                const bool desc = ((i & size) == 0);
                if ((a < b) == desc) { s[i] = b; s[j] = a; }
            }
            __syncthreads();
        }
    }

    // position k (0-based, descending) has rank k-1 -> weight exp(-(k-1)*alpha)
    float p = 0.f;
    for (int k = tid; k < NN; k += 256) {
        const float w = __expf(-ALPHA * (float)(k - 1));
        const float l = fmaxf(-__logf(s[k]), 0.0f);
        p += w * l;
    }
    red[tid] = p;
    __syncthreads();
    for (int off = 128; off > 0; off >>= 1) {
        if (tid < off) red[tid] += red[tid + off];
        __syncthreads();
    }
    if (tid == 0) rowsum[row] = red[0];
}

// ---------------- Kernel 4: deterministic final reduction --------------------
__global__ __launch_bounds__(256) void finalize_kernel(const float* __restrict__ rowsum,
                                                       float* __restrict__ out, int N) {
    __shared__ float red[256];
    float p = 0.f;
    for (int k = threadIdx.x; k < N; k += 256) p += rowsum[k];
    red[threadIdx.x] = p;
    __syncthreads();
    for (int off = 128; off > 0; off >>= 1) {
        if (threadIdx.x < off) red[threadIdx.x] += red[threadIdx.x + off];
        __syncthreads();
    }
    if (threadIdx.x == 0) out[0] = red[0] / ((float)N * (float)N);
}

extern "C" void kernel_launch(void* const* d_in, const int* in_sizes, int n_in,
                              void* d_out, int out_size, void* d_ws, size_t ws_size,
                              hipStream_t stream) {
    const float* e = (const float*)d_in[0];
    const int C = C_DIM;
    const int N = in_sizes[0] / C;   // 8192

    // Workspace layout: rowsum[N] | xn[N*C] | sim[N*N]
    char* ws = (char*)d_ws;
    float* rowsum = (float*)ws;
    size_t xnOff  = ((size_t)N * 4 + 255) & ~(size_t)255;
    float* xn     = (float*)(ws + xnOff);
    float* sim    = (float*)(ws + xnOff + (size_t)N * C * 4);

    rownorm_kernel<<<N, 256, 0, stream>>>(e, xn, C);

    dim3 grid(N / BT, N / BT);
    simgemm_kernel<<<grid, 256, 0, stream>>>(xn, sim, N, C);

    rowsort_kernel<<<N, 256, 0, stream>>>(sim, rowsum, N);

    finalize_kernel<<<1, 256, 0, stream>>>(rowsum, (float*)d_out, N);
}